// RecurrentKV_85950885528326
// MI455X (gfx1250) — compile-verified
//
#include <hip/hip_runtime.h>

// Recurrent KV (linear attention) scan, chunked into dense WMMA GEMMs.
// B=4, H=8, T=1024, D=64, decay=0.95.  f32 end-to-end via V_WMMA_F32_16X16X4_F32.

#define BH      32          // B*H
#define T       1024
#define DD      64          // head dim
#define CHUNK   64
#define NCHUNK  (T / CHUNK)
#define EB      16          // state columns (e) handled per block
#define NE      (DD / EB)   // e-slices -> 4
#define QSTR    68          // LDS pitch for 64-wide arrays (bank-friendly)
#define VSTR    24          // LDS pitch for 16-wide arrays

#define L2DECAY     (-0.07400058144377693f)  // log2(0.95)
#define DECAY_CHUNK (0.0375241347f)          // 0.95^64

typedef __attribute__((ext_vector_type(2))) float v2f;
typedef __attribute__((ext_vector_type(8))) float v8f;

__device__ __forceinline__ v8f wmma_f32(v2f a, v2f b, v8f c) {
    return __builtin_amdgcn_wmma_f32_16x16x4_f32(
        /*neg_a=*/false, a, /*neg_b=*/false, b,
        /*c_mod=*/(short)0, c, /*reuse_a=*/false, /*reuse_b=*/false);
}

__global__ __launch_bounds__(128)
void rkv_chunked_wmma(const float* __restrict__ q,
                      const float* __restrict__ k,
                      const float* __restrict__ v,
                      float* __restrict__ out)
{
    __shared__ float sQ[CHUNK][QSTR];   // Qd[t][d] = decay^(t+1) * q
    __shared__ float sK[CHUNK][QSTR];   // Kd[s][d] = decay^-(s+1) * k
    __shared__ float sA[CHUNK][QSTR];   // masked attention scores
    __shared__ float sV[CHUNK][VSTR];   // V slice [t][e_local]
    __shared__ float sS[DD][VSTR];      // state slice [d][e_local]

    const int tid  = threadIdx.x;
    const int lane = tid & 31;
    const int wave = tid >> 5;          // 0..3
    const int lr   = lane & 15;         // fragment row/col select
    const int half = lane >> 4;         // lane-half -> K/row offset
    const int m0   = wave * 16;         // this wave's 16-row band

    const int bh = blockIdx.x / NE;
    const int e0 = (blockIdx.x % NE) * EB;

    const float* qbh = q + (size_t)bh * T * DD;
    const float* kbh = k + (size_t)bh * T * DD;
    const float* vbh = v + (size_t)bh * T * DD;
    float*       obh = out + (size_t)bh * T * DD;

    // zero the state slice
    for (int i = tid; i < DD * EB; i += 128)
        sS[i / EB][i % EB] = 0.0f;

    for (int c = 0; c < NCHUNK; ++c) {
        __syncthreads();   // prior chunk's consumers of sQ/sK/sV done
        const int tbase = c * CHUNK;

        // ---- load chunk, applying decay pre-scales (float4, coalesced) ----
        for (int i = tid; i < CHUNK * (DD / 4); i += 128) {
            const int row = i / (DD / 4);
            const int c4  = (i % (DD / 4)) * 4;
            const float4 qv = *(const float4*)(qbh + (size_t)(tbase + row) * DD + c4);
            const float4 kv = *(const float4*)(kbh + (size_t)(tbase + row) * DD + c4);
            const float qs = __builtin_amdgcn_exp2f((float)(row + 1) * L2DECAY);     // decay^(t+1)
            const float ks = __builtin_amdgcn_exp2f((float)(-(row + 1)) * L2DECAY);  // decay^-(s+1)
            sQ[row][c4 + 0] = qv.x * qs; sQ[row][c4 + 1] = qv.y * qs;
            sQ[row][c4 + 2] = qv.z * qs; sQ[row][c4 + 3] = qv.w * qs;
            sK[row][c4 + 0] = kv.x * ks; sK[row][c4 + 1] = kv.y * ks;
            sK[row][c4 + 2] = kv.z * ks; sK[row][c4 + 3] = kv.w * ks;
        }
        for (int i = tid; i < CHUNK * (EB / 4); i += 128) {
            const int row = i / (EB / 4);
            const int c4  = (i % (EB / 4)) * 4;
            const float4 vv = *(const float4*)(vbh + (size_t)(tbase + row) * DD + e0 + c4);
            sV[row][c4 + 0] = vv.x; sV[row][c4 + 1] = vv.y;
            sV[row][c4 + 2] = vv.z; sV[row][c4 + 3] = vv.w;
        }
        __syncthreads();

        // ---- A = tril(Qd @ Kd^T), 64x64: wave does its 16-row band x 4 tiles ----
        for (int nb = 0; nb < 4; ++nb) {
            const int n0 = nb * 16;
            v8f acc = {};
            #pragma unroll
            for (int k0 = 0; k0 < DD; k0 += 4) {
                v2f a, b;
                a[0] = sQ[m0 + lr][k0 + 2 * half + 0];
                a[1] = sQ[m0 + lr][k0 + 2 * half + 1];
                b[0] = sK[n0 + lr][k0 + 2 * half + 0];   // B = Kd^T: col n is Kd row n
                b[1] = sK[n0 + lr][k0 + 2 * half + 1];
                acc = wmma_f32(a, b, acc);
            }
            #pragma unroll
            for (int r = 0; r < 8; ++r) {
                const int t = m0 + r + 8 * half;   // C/D layout row
                const int s = n0 + lr;             // C/D layout col
                sA[t][s] = (s <= t) ? acc[r] : 0.0f;  // causal decay mask
            }
        }
        __syncthreads();

        // ---- O = A @ V + Qd @ S_prev  (16-wide e-slice: one tile per wave) ----
        {
            v8f acc = {};
            #pragma unroll
            for (int k0 = 0; k0 < CHUNK; k0 += 4) {   // A @ V over s
                v2f a, b;
                a[0] = sA[m0 + lr][k0 + 2 * half + 0];
                a[1] = sA[m0 + lr][k0 + 2 * half + 1];
                b[0] = sV[k0 + 2 * half + 0][lr];
                b[1] = sV[k0 + 2 * half + 1][lr];
                acc = wmma_f32(a, b, acc);
            }
            #pragma unroll
            for (int k0 = 0; k0 < DD; k0 += 4) {      // Qd @ S_prev over d
                v2f a, b;
                a[0] = sQ[m0 + lr][k0 + 2 * half + 0];
                a[1] = sQ[m0 + lr][k0 + 2 * half + 1];
                b[0] = sS[k0 + 2 * half + 0][lr];
                b[1] = sS[k0 + 2 * half + 1][lr];
                acc = wmma_f32(a, b, acc);
            }
            #pragma unroll
            for (int r = 0; r < 8; ++r) {
                const int t = tbase + m0 + r + 8 * half;
                const int e = e0 + lr;
                obh[(size_t)t * DD + e] = acc[r];
            }
        }
        __syncthreads();   // all reads of S_prev complete before update

        // ---- S = decay^C * (S + Kd^T @ V)  (wave updates its own d-band) ----
        {
            v8f acc;
            #pragma unroll
            for (int r = 0; r < 8; ++r)
                acc[r] = sS[m0 + r + 8 * half][lr];   // C operand = current state
            #pragma unroll
            for (int k0 = 0; k0 < CHUNK; k0 += 4) {   // sum over s
                v2f a, b;
                a[0] = sK[k0 + 2 * half + 0][m0 + lr];  // A[d][s] = Kd[s][d]
                a[1] = sK[k0 + 2 * half + 1][m0 + lr];
                b[0] = sV[k0 + 2 * half + 0][lr];
                b[1] = sV[k0 + 2 * half + 1][lr];
                acc = wmma_f32(a, b, acc);
            }
            #pragma unroll
            for (int r = 0; r < 8; ++r)
                sS[m0 + r + 8 * half][lr] = DECAY_CHUNK * acc[r];
        }
    }
}

extern "C" void kernel_launch(void* const* d_in, const int* in_sizes, int n_in,
                              void* d_out, int out_size, void* d_ws, size_t ws_size,
                              hipStream_t stream) {
    (void)in_sizes; (void)n_in; (void)d_ws; (void)ws_size; (void)out_size;
    const float* q = (const float*)d_in[0];
    const float* k = (const float*)d_in[1];
    const float* v = (const float*)d_in[2];
    float* out = (float*)d_out;
    rkv_chunked_wmma<<<dim3(BH * NE), dim3(128), 0, stream>>>(q, k, v, out);
}